// PIClauseInferModule_28260884808447
// MI455X (gfx1250) — compile-verified
//
#include <hip/hip_runtime.h>
#include <stdint.h>

// ---------------------------------------------------------------------------
// PIClauseInferModule: differentiable forward chaining (soft-OR logsumexp)
// C=4 clauses, B=128 batch, G=2048 ground atoms, S=32 substitutions, L=3 body
// ---------------------------------------------------------------------------

#define GAMMA_F 0.01f
#define INV_GAMMA_F 100.0f

constexpr int C_ = 4;
constexpr int B_ = 128;
constexpr int G_ = 2048;
constexpr int S_ = 32;
constexpr int N_ = C_ * B_ * G_;     // 1,048,576

constexpr int BROWS  = 8;            // R rows staged per block (64KB LDS tile)
constexpr int GCHUNK = 512;          // g values per block
constexpr int NBB    = B_ / BROWS;   // 16
constexpr int NGC    = G_ / GCHUNK;  // 4

typedef unsigned int uv4 __attribute__((ext_vector_type(4)));
typedef int          iv8 __attribute__((ext_vector_type(8)));
typedef int          iv4 __attribute__((ext_vector_type(4)));

// ---------------------------------------------------------------------------
// Init: R[c,b,g] = x[b,g]; zero the atomic-max scalar slots.
// ---------------------------------------------------------------------------
__global__ void init_kernel(const float* __restrict__ x, float* __restrict__ R,
                            unsigned* __restrict__ scal) {
  int i = blockIdx.x * blockDim.x + threadIdx.x;
  if (blockIdx.x == 0 && threadIdx.x < 8) scal[threadIdx.x] = 0u;
  if (i < N_) {
    int g = i & (G_ - 1);
    int b = (i >> 11) & (B_ - 1);
    R[i] = x[b * G_ + g];
  }
}

// ---------------------------------------------------------------------------
// Kernel A: clause_eval. One block = (clause c, 8 batch rows, 512 g values).
//   - One 2-D TDM DMA stages R[c, b0:b0+8, :] (64KB) into LDS
//   - lane = substitution s (S==32 == wave32); index triple loaded once,
//     gathered against all 8 staged rows (8x index-traffic amortization)
//   - butterfly logsumexp per (b,g), block max -> atomicMax into m1
// ---------------------------------------------------------------------------
__global__ __launch_bounds__(256) void clause_eval_kernel(
    const float* __restrict__ R, const int* __restrict__ I,
    float* __restrict__ Q, unsigned* __restrict__ m1slot) {
  __shared__ float tile[BROWS][G_];   // exactly 64KB; reused for reduction

  const int bx   = blockIdx.x;        // = ((c*NBB + bbi)*NGC + gci)
  const int gci  = bx & (NGC - 1);
  const int bbi  = (bx >> 2) & (NBB - 1);
  const int c    = bx >> 6;
  const int tid  = threadIdx.x;
  const int lane = tid & 31;
  const int wave = tid >> 5;

  const int b0    = bbi * BROWS;
  const int gbase = gci * GCHUNK;
  const long long qbase = (long long)(c * B_ + b0) * G_;

  // --- TDM: 2-D tile (8 rows x 2048 floats) of R into LDS, one DMA ---------
  if (wave == 0) {
    unsigned long long gaddr = (unsigned long long)(uintptr_t)(R + qbase);
    unsigned ldsoff = (unsigned)(uintptr_t)&tile[0][0];

    uv4 g0;
    g0.x = 1u;                                        // count=1 (valid user D#)
    g0.y = ldsoff;                                    // lds_addr
    g0.z = (unsigned)(gaddr & 0xFFFFFFFFull);         // global_addr[31:0]
    g0.w = (unsigned)(gaddr >> 32) | (2u << 30);      // global_addr[56:32], type=2

    iv8 g1;
    g1[0] = (int)(2u << 16);                          // data_size=2 (4 bytes)
    g1[1] = (int)((unsigned)G_ << 16);                // tensor_dim0[15:0] @ [63:48]
    g1[2] = (int)((unsigned)BROWS << 16);             // dim0 hi=0, tensor_dim1=8
    g1[3] = (int)((unsigned)G_ << 16);                // dim1 hi=0, tile_dim0=2048
    g1[4] = BROWS;                                    // tile_dim1=8, tile_dim2=0
    g1[5] = G_;                                       // tensor_dim0_stride[31:0]
    g1[6] = 0;                                        // stride hi bits = 0
    g1[7] = 0;
    iv4 g2 = {};
    iv4 g3 = {};
    iv8 g4 = {};
    __builtin_amdgcn_tensor_load_to_lds(g0, g1, g2, g3, g4, /*cpol=*/0);
    __builtin_amdgcn_s_wait_tensorcnt(0);
  }
  __syncthreads();

  // --- soft-OR over substitutions; one index triple feeds 8 rows -----------
  float localmax = 0.0f;

  for (int gi = wave; gi < GCHUNK; gi += 8) {
    const int g = gbase + gi;
    const int* ip = I + (((long long)(c * G_ + g) * S_) + lane) * 3;
    __builtin_prefetch(ip + 8 * S_ * 3, 0, 0);  // next gi for this wave (+3KB)
    const int i0 = ip[0];
    const int i1 = ip[1];
    const int i2 = ip[2];

    #pragma unroll
    for (int b = 0; b < BROWS; ++b) {
      float body = tile[b][i0] * tile[b][i1] * tile[b][i2];

      float m = body;                      // wave32 butterfly max
      #pragma unroll
      for (int off = 16; off > 0; off >>= 1)
        m = fmaxf(m, __shfl_xor(m, off));
      float e = __expf((body - m) * INV_GAMMA_F);
      #pragma unroll
      for (int off = 16; off > 0; off >>= 1)
        e += __shfl_xor(e, off);

      float lse = m + GAMMA_F * __logf(e); // gamma * logsumexp(body/gamma)
      if (lane == 0) Q[qbase + (long long)b * G_ + g] = lse;
      localmax = fmaxf(localmax, lse);     // identical across lanes
    }
  }

  // --- block max -> global atomic max (values >= 0: uint-pun is exact) -----
  __syncthreads();                         // tile no longer needed; reuse it
  float* wred = &tile[0][0];
  #pragma unroll
  for (int off = 16; off > 0; off >>= 1)
    localmax = fmaxf(localmax, __shfl_xor(localmax, off));
  if (lane == 0) wred[wave] = localmax;
  __syncthreads();
  if (tid == 0) {
    float bm = wred[0];
    #pragma unroll
    for (int w = 1; w < 8; ++w) bm = fmaxf(bm, wred[w]);
    atomicMax(m1slot, __float_as_uint(bm));
  }
}

// ---------------------------------------------------------------------------
// Kernel B: normalize Q by m1 (if >1), pairwise soft-OR with R, write lse2
// back into Q, atomicMax of lse2 into m2.
// ---------------------------------------------------------------------------
__global__ __launch_bounds__(256) void pair_softor_kernel(
    const float* __restrict__ R, float* __restrict__ Q,
    const unsigned* __restrict__ m1slot, unsigned* __restrict__ m2slot) {
  __shared__ float wred[8];
  const int i    = blockIdx.x * 256 + threadIdx.x;
  const int lane = threadIdx.x & 31;
  const int wave = threadIdx.x >> 5;

  const float m1 = __uint_as_float(*m1slot);
  float lse2 = 0.0f;
  if (i < N_) {
    float q = Q[i];
    if (m1 > 1.0f) q /= m1;
    float r = R[i];
    float t = fmaxf(r, q);
    lse2 = t + GAMMA_F * __logf(__expf((r - t) * INV_GAMMA_F) +
                                __expf((q - t) * INV_GAMMA_F));
    Q[i] = lse2;
  }

  float lm = lse2;
  #pragma unroll
  for (int off = 16; off > 0; off >>= 1)
    lm = fmaxf(lm, __shfl_xor(lm, off));
  if (lane == 0) wred[wave] = lm;
  __syncthreads();
  if (threadIdx.x == 0) {
    float bm = wred[0];
    #pragma unroll
    for (int w = 1; w < 8; ++w) bm = fmaxf(bm, wred[w]);
    atomicMax(m2slot, __float_as_uint(bm));
  }
}

// ---------------------------------------------------------------------------
// Kernel C: R = (m2 > 1) ? lse2/m2 : lse2
// ---------------------------------------------------------------------------
__global__ void normalize_kernel(const float* __restrict__ Q, float* __restrict__ R,
                                 const unsigned* __restrict__ mslot) {
  int i = blockIdx.x * blockDim.x + threadIdx.x;
  if (i >= N_) return;
  float m = __uint_as_float(*mslot);
  float v = Q[i];
  R[i] = (m > 1.0f) ? (v / m) : v;
}

// ---------------------------------------------------------------------------
extern "C" void kernel_launch(void* const* d_in, const int* in_sizes, int n_in,
                              void* d_out, int out_size, void* d_ws, size_t ws_size,
                              hipStream_t stream) {
  const float* x = (const float*)d_in[0];  // [B,G]
  const int*   I = (const int*)d_in[1];    // [C,G,S,L]
  float*       R = (float*)d_out;          // [C,B,G] (doubles as the R buffer)

  float*    Q    = (float*)d_ws;                                          // [C,B,G]
  unsigned* scal = (unsigned*)((char*)d_ws + (size_t)N_ * sizeof(float)); // 8 slots

  init_kernel<<<(N_ + 255) / 256, 256, 0, stream>>>(x, R, scal);

  for (int step = 0; step < 3; ++step) {
    unsigned* m1 = scal + 2 * step;
    unsigned* m2 = m1 + 1;
    clause_eval_kernel<<<C_ * NBB * NGC, 256, 0, stream>>>(R, I, Q, m1);
    pair_softor_kernel<<<(N_ + 255) / 256, 256, 0, stream>>>(R, Q, m1, m2);
    normalize_kernel<<<(N_ + 255) / 256, 256, 0, stream>>>(Q, R, m2);
  }
}